// TTEmbedding_61211873903090
// MI455X (gfx1250) — compile-verified
//
#include <hip/hip_runtime.h>

typedef __attribute__((ext_vector_type(2))) float v2f;
typedef __attribute__((ext_vector_type(4))) float v4f;
typedef __attribute__((ext_vector_type(8))) float v8f;

// TT-embedding: VOC_Q=(32,40,40), EMB_Q=(8,8,16), ranks 8,8.
// g1: [32][8][8]   (i,a,p)
// g2: [8][40][8][8](p,j,b,q)
// g3: [8][40][16]  (q,k,c)
// out row for token n=(i,j,k): out[(a*8+b)*16+c] = sum_{p,q} g1*g2*g3

__global__ __launch_bounds__(256) void tt_embed_wmma_kernel(
    const int*   __restrict__ x,
    const float* __restrict__ g1,
    const float* __restrict__ g2,
    const float* __restrict__ g3,
    float*       __restrict__ out,
    int tokens)
{
    // Per-wave LDS: A[64] | B[512] | C[128] | T[512] | O[1024] = 2240 floats
    __shared__ float lds[8][2240];

    const int lane  = threadIdx.x & 31;
    const int wave  = threadIdx.x >> 5;
    const int token = blockIdx.x * 8 + wave;          // one wave per token
    // Tail clamp: a clamped wave recomputes token tokens-1 and stores the
    // identical bytes again (benign duplicate write) -> no EXEC divergence.
    const int tok   = token < tokens ? token : tokens - 1;

    const int n   = x[tok];
    const int i   = n / 1600;        // 40*40
    const int rem = n % 1600;
    const int j   = rem / 40;
    const int k   = rem % 40;

    float* A = &lds[wave][0];      // A[a*8+p]
    float* B = &lds[wave][64];     // B[p*64 + b*8 + q]
    float* C = &lds[wave][576];    // C[q*16 + c]
    float* T = &lds[wave][704];    // T[(a*8+b)*8 + q]
    float* O = &lds[wave][1216];   // O[1024] output staging

    // ---- cooperative loads (g1/g2/g3 are tiny -> L2/WGP$-resident) ----
    {   // g1[i] : 64 contiguous floats
        const float* src = g1 + i * 64;
        A[lane]      = src[lane];
        A[lane + 32] = src[lane + 32];
    }
    #pragma unroll
    for (int t = 0; t < 16; ++t) {   // g2[:,j,:,:] : 8 blocks of 64 floats
        int idx = t * 32 + lane;
        int p   = idx >> 6;
        int bq  = idx & 63;
        B[idx]  = g2[(p * 40 + j) * 64 + bq];
    }
    #pragma unroll
    for (int t = 0; t < 4; ++t) {    // g3[:,k,:] : 8 blocks of 16 floats
        int idx = t * 32 + lane;
        int q   = idx >> 4;
        int c   = idx & 15;
        C[idx]  = g3[(q * 40 + k) * 16 + c];
    }
    __syncthreads();

    // ---- step 1 (VALU): T[ab,q] = sum_p A[a,p] * B[p,b,q]  (64x8) ----
    #pragma unroll
    for (int t = 0; t < 16; ++t) {
        int idx = t * 32 + lane;     // 0..511, consecutive lanes -> no bank conflict
        int row = idx >> 3;          // a*8+b
        int q   = idx & 7;
        int a   = row >> 3;
        int b   = row & 7;
        float acc = 0.0f;
        #pragma unroll
        for (int p = 0; p < 8; ++p)
            acc += A[a * 8 + p] * B[p * 64 + b * 8 + q];
        T[row * 8 + q] = acc;
    }
    __syncthreads();

    // ---- step 2 (WMMA f32 16x16x4): out[64x16] = T[64x8] @ C[8x16] ----
    const int   hs   = lane >> 4;          // half-wave select (0/1)
    const int   ln   = lane & 15;
    const float mask = (n != 0) ? 1.0f : 0.0f;   // PADDING_IDX = 0

    v8f accs[4];
    #pragma unroll
    for (int mtile = 0; mtile < 4; ++mtile) {
        v8f acc = {};
        #pragma unroll
        for (int ks = 0; ks < 2; ++ks) {
            // A fragment 16x4 f32: lanes 0-15 hold K={0,1}, 16-31 hold K={2,3}
            const int kbase = ks * 4 + hs * 2;
            const int rowM  = mtile * 16 + ln;
            v2f af;
            af.x = T[rowM * 8 + kbase];
            af.y = T[rowM * 8 + kbase + 1];
            // B fragment 4x16 f32: same K split, N = lane%16
            v2f bf;
            bf.x = C[kbase * 16 + ln];
            bf.y = C[(kbase + 1) * 16 + ln];
            acc = __builtin_amdgcn_wmma_f32_16x16x4_f32(
                false, af, false, bf, (short)0, acc, false, false);
        }
        accs[mtile] = acc;
    }

    // ---- repack D fragments -> LDS -> contiguous b128 stores ----
    // D layout: VGPR v holds (M = 16*mtile + v + hs*8, N = ln); flat output
    // offset = M*16 + N = mtile*256 + (v + hs*8)*16 + ln.
    #pragma unroll
    for (int mtile = 0; mtile < 4; ++mtile) {
        #pragma unroll
        for (int v = 0; v < 8; ++v)
            O[mtile * 256 + (v + hs * 8) * 16 + ln] = accs[mtile][v] * mask;
    }
    __syncthreads();   // orders per-wave DS writes before cross-lane reads

    // 1024 floats = 256 float4; each lane stores 8 float4 -> each
    // global_store_b128 covers a contiguous 512B span across the wave.
    const v4f* O4  = (const v4f*)O;
    v4f* dst4 = (v4f*)(out + (size_t)tok * 1024);
    #pragma unroll
    for (int t = 0; t < 8; ++t)
        dst4[t * 32 + lane] = O4[t * 32 + lane];
}

extern "C" void kernel_launch(void* const* d_in, const int* in_sizes, int n_in,
                              void* d_out, int out_size, void* d_ws, size_t ws_size,
                              hipStream_t stream) {
    const int*   x  = (const int*)  d_in[0];
    const float* g1 = (const float*)d_in[1];
    const float* g2 = (const float*)d_in[2];
    const float* g3 = (const float*)d_in[3];
    float* out = (float*)d_out;

    int tokens = in_sizes[0];            // 8*2048 = 16384
    int blocks = (tokens + 7) / 8;       // one wave (32 lanes) per token
    tt_embed_wmma_kernel<<<blocks, 256, 0, stream>>>(x, g1, g2, g3, out, tokens);
}